// CustomNeighborSearch_37838661877946
// MI455X (gfx1250) — compile-verified
//
#include <hip/hip_runtime.h>
#include <hip/hip_bf16.h>

typedef __attribute__((ext_vector_type(2))) float v2f;
typedef __attribute__((ext_vector_type(8))) float v8f;

#define MQ 12288
#define ND 12288

// One wave computes one 16x16 tile of the [MQ, ND] distance matrix with a
// single V_WMMA_F32_16X16X4_F32:  D = A(16x4) x B(4x16) + C(16x16)
//   A[m] = [-2qx, -2qy, -2qz, 1],  B[:,n] = [dx, dy, dz, |d|^2],  C[m,n] = |q_m|^2
// Output writes (~1.21 GB total, 6x the 192 MB L2) are tagged non-temporal so
// the streaming store path doesn't evict the cache-resident 300 KB of inputs.
__global__ __launch_bounds__(256) void nbr_tile_kernel(
    const float* __restrict__ data,     // [ND][3]
    const float* __restrict__ queries,  // [MQ][3]
    const float* __restrict__ radius,   // [1]
    float* __restrict__ mask_out,       // [MQ][ND] (0.0/1.0)
    float* __restrict__ w_out,          // [MQ][ND] (masked d2)
    int* __restrict__ counts)           // [MQ]
{
    const int lane  = threadIdx.x & 31;
    const int wave  = threadIdx.x >> 5;
    const int tile_n = blockIdx.x * 8 + wave;   // 768 tiles along N
    const int tile_m = blockIdx.y;              // 768 tiles along M

    const int half = lane >> 4;     // 0: lanes 0-15 (K=0,1), 1: lanes 16-31 (K=2,3)
    const int l16  = lane & 15;

    const int m = tile_m * 16 + l16;
    const int n = tile_n * 16 + l16;

    // All lanes load their query row and data column (both halves redundant;
    // 300 KB total input lives in WGP$/L2, so this is free and keeps EXEC full).
    const float qx = queries[3 * m + 0];
    const float qy = queries[3 * m + 1];
    const float qz = queries[3 * m + 2];
    const float dx = data[3 * n + 0];
    const float dy = data[3 * n + 1];
    const float dz = data[3 * n + 2];
    const float qn = qx * qx + qy * qy + qz * qz;   // |q_m|^2
    const float dn = dx * dx + dy * dy + dz * dz;   // |d_n|^2

    // A fragment (16x4 f32): lanes 0-15 -> {A[m][0], A[m][1]}, lanes 16-31 -> {A[m][2], A[m][3]}
    v2f a;
    a.x = half ? (-2.0f * qz) : (-2.0f * qx);
    a.y = half ? 1.0f         : (-2.0f * qy);
    // B fragment (4x16 f32): lanes 0-15 -> {B[0][n], B[1][n]}, lanes 16-31 -> {B[2][n], B[3][n]}
    v2f b;
    b.x = half ? dz : dx;
    b.y = half ? dn : dy;
    // C fragment: C[row][*] = |q_row|^2 ; VGPR j holds rows j (lo lanes) / j+8 (hi lanes)
    v8f c;
#pragma unroll
    for (int j = 0; j < 8; ++j) {
        c[j] = __shfl(qn, half ? (j + 8) : j, 32);
    }

    v8f dres = __builtin_amdgcn_wmma_f32_16x16x4_f32(
        /*neg_a=*/false, a, /*neg_b=*/false, b,
        /*c_mod=*/(short)0, c, /*reuse_a=*/false, /*reuse_b=*/false);

    const float r  = radius[0];
    const float r2 = r * r;

    unsigned bals[8];
#pragma unroll
    for (int j = 0; j < 8; ++j) {
        float d2raw = dres[j];
        float d2 = (d2raw < 0.0f) ? 0.0f : d2raw;   // clamp (inputs finite -> no NaN)
        bool inr = d2 <= r2;
        bals[j] = (unsigned)__ballot(inr);  // bits 0-15: row j, bits 16-31: row j+8
        size_t idx = (size_t)(tile_m * 16 + j + 8 * half) * (size_t)ND + (size_t)n;
        __builtin_nontemporal_store(inr ? 1.0f : 0.0f, &mask_out[idx]);
        __builtin_nontemporal_store(inr ? d2 : 0.0f,   &w_out[idx]);
    }

    // Per-row neighbor counts for this tile: lane t (t<16) owns row tile_m*16+t.
    if (lane < 16) {
        unsigned bl = bals[lane & 7];
        int cnt = (lane < 8) ? __popc(bl & 0xFFFFu) : __popc(bl >> 16);
        atomicAdd(&counts[tile_m * 16 + lane], cnt);
    }
}

__global__ void zero_counts_kernel(int* __restrict__ counts, int nrows) {
    int i = blockIdx.x * blockDim.x + threadIdx.x;
    if (i < nrows) counts[i] = 0;
}

// Exclusive prefix sum of counts[MQ] -> row_splits[MQ+1] (written as floats).
__global__ __launch_bounds__(256) void scan_kernel(const int* __restrict__ counts,
                                                   float* __restrict__ row_splits) {
    __shared__ int partials[256];
    const int tid   = threadIdx.x;
    const int chunk = MQ / 256;          // 48
    const int base  = tid * chunk;

    int s = 0;
    for (int i = 0; i < chunk; ++i) s += counts[base + i];
    partials[tid] = s;
    __syncthreads();

    if (tid == 0) {
        int run = 0;
        for (int j = 0; j < 256; ++j) {
            int t = partials[j];
            partials[j] = run;
            run += t;
        }
    }
    __syncthreads();

    int run = partials[tid];
    for (int i = 0; i < chunk; ++i) {
        row_splits[base + i] = (float)run;   // row_splits[k] = sum counts[0..k-1]
        run += counts[base + i];
    }
    if (tid == 255) row_splits[MQ] = (float)run;  // total neighbor count
}

extern "C" void kernel_launch(void* const* d_in, const int* in_sizes, int n_in,
                              void* d_out, int out_size, void* d_ws, size_t ws_size,
                              hipStream_t stream) {
    const float* data    = (const float*)d_in[0];
    const float* queries = (const float*)d_in[1];
    const float* radius  = (const float*)d_in[2];

    float* out        = (float*)d_out;
    float* row_splits = out;                                   // [MQ+1]
    float* mask_out   = out + (MQ + 1);                        // [MQ*ND]
    float* w_out      = mask_out + (size_t)MQ * (size_t)ND;    // [MQ*ND]
    int*   counts     = (int*)d_ws;                            // [MQ]

    hipLaunchKernelGGL(zero_counts_kernel, dim3((MQ + 255) / 256), dim3(256), 0, stream,
                       counts, MQ);

    dim3 grid(ND / 16 / 8, MQ / 16);  // (96, 768): 8 tiles (waves) per block
    hipLaunchKernelGGL(nbr_tile_kernel, grid, dim3(256), 0, stream,
                       data, queries, radius, mask_out, w_out, counts);

    hipLaunchKernelGGL(scan_kernel, dim3(1), dim3(256), 0, stream,
                       counts, row_splits);
}